// Point2Point_Loss_76166950027360
// MI455X (gfx1250) — compile-verified
//
#include <hip/hip_runtime.h>

// ---------------------------------------------------------------------------
// Problem constants (match reference: SUBDIVISIONS=7, BATCH=16)
// ---------------------------------------------------------------------------
constexpr int BATCH = 16;
constexpr int SUB   = 7;
constexpr int BH    = 1 << SUB;        // 128
constexpr int GY    = 5 * BH;          // 640
constexpr int GX    = 2 * BH;          // 256
constexpr int NGRID = GY * GX;         // 163840
constexpr int NV    = NGRID + 2;       // 163842 vertices
constexpr int SZ    = BATCH * NV * 3;  // floats per (B,NV,3) buffer = 7,864,416

constexpr int THREADS      = 256;
constexpr int NBLOCKS_LOSS = (BATCH * NV + THREADS - 1) / THREADS;   // 10241
constexpr int NB64         = ((NBLOCKS_LOSS + 63) / 64) * 64;        // 10304

typedef __attribute__((ext_vector_type(2))) float v2f;
typedef __attribute__((ext_vector_type(8))) float v8f;

// ---------------------------------------------------------------------------
// Zero scatter targets (vn, nbr_sum) and partial-sum arrays (incl. pad)
// ---------------------------------------------------------------------------
__global__ __launch_bounds__(THREADS)
void zero_kernel(float* __restrict__ p, int n) {
    int i = blockIdx.x * THREADS + threadIdx.x;
    if (i < n) p[i] = 0.0f;
}

// ---------------------------------------------------------------------------
// Build v[(b*NV + i)*3 + c] from inputs (B,3,GY,GX); poles = 5-sample means.
// ---------------------------------------------------------------------------
__global__ __launch_bounds__(THREADS)
void build_v_kernel(const float* __restrict__ in, float* __restrict__ v) {
    int idx = blockIdx.x * THREADS + threadIdx.x;     // over BATCH*NV
    if (idx >= BATCH * NV) return;
    int b = idx / NV;
    int i = idx - b * NV;
    const float* src = in + (size_t)b * 3 * NGRID;
    float out[3];
    if (i < NGRID) {
#pragma unroll
        for (int c = 0; c < 3; ++c)
            out[c] = src[(size_t)c * NGRID + i];
    } else if (i == NGRID) {            // top pole: rows k*BH (k=0..4), col 0
#pragma unroll
        for (int c = 0; c < 3; ++c) {
            float s = 0.0f;
            for (int k = 0; k < 5; ++k)
                s += src[(size_t)c * NGRID + (size_t)(k * BH) * GX];
            out[c] = s * 0.2f;
        }
    } else {                            // bottom pole: rows k*BH-1 (k=1..5), col GX-1
#pragma unroll
        for (int c = 0; c < 3; ++c) {
            float s = 0.0f;
            for (int k = 1; k <= 5; ++k)
                s += src[(size_t)c * NGRID + (size_t)(k * BH - 1) * GX + (GX - 1)];
            out[c] = s * 0.2f;
        }
    }
    float* dst = v + (size_t)idx * 3;
    dst[0] = out[0]; dst[1] = out[1]; dst[2] = out[2];
}

// ---------------------------------------------------------------------------
// Face normals: fn = cross(v1-v0, v2-v0), scatter-add into vn at f0,f1,f2.
// ---------------------------------------------------------------------------
__global__ __launch_bounds__(THREADS)
void face_kernel(const int* __restrict__ faces, const float* __restrict__ v,
                 float* __restrict__ vn, int nf) {
    int idx = blockIdx.x * THREADS + threadIdx.x;     // over BATCH*nf
    if (idx >= BATCH * nf) return;
    int b = idx / nf;
    int f = idx - b * nf;
    int f0 = faces[f * 3 + 0];
    int f1 = faces[f * 3 + 1];
    int f2 = faces[f * 3 + 2];
    const float* vb = v + (size_t)b * NV * 3;
    const float* p0 = vb + (size_t)f0 * 3;
    const float* p1 = vb + (size_t)f1 * 3;
    const float* p2 = vb + (size_t)f2 * 3;
    float ax = p1[0] - p0[0], ay = p1[1] - p0[1], az = p1[2] - p0[2];
    float bx = p2[0] - p0[0], by = p2[1] - p0[1], bz = p2[2] - p0[2];
    float nx = ay * bz - az * by;
    float ny = az * bx - ax * bz;
    float nz = ax * by - ay * bx;
    float* vnb = vn + (size_t)b * NV * 3;
    atomicAdd(&vnb[(size_t)f0 * 3 + 0], nx);
    atomicAdd(&vnb[(size_t)f0 * 3 + 1], ny);
    atomicAdd(&vnb[(size_t)f0 * 3 + 2], nz);
    atomicAdd(&vnb[(size_t)f1 * 3 + 0], nx);
    atomicAdd(&vnb[(size_t)f1 * 3 + 1], ny);
    atomicAdd(&vnb[(size_t)f1 * 3 + 2], nz);
    atomicAdd(&vnb[(size_t)f2 * 3 + 0], nx);
    atomicAdd(&vnb[(size_t)f2 * 3 + 1], ny);
    atomicAdd(&vnb[(size_t)f2 * 3 + 2], nz);
}

// ---------------------------------------------------------------------------
// Edge scatter: nbr_sum[b,dst] += v[b,src]
// ---------------------------------------------------------------------------
__global__ __launch_bounds__(THREADS)
void edge_kernel(const int* __restrict__ esrc, const int* __restrict__ edst,
                 const float* __restrict__ v, float* __restrict__ nbr, int ne) {
    int idx = blockIdx.x * THREADS + threadIdx.x;     // over BATCH*ne
    if (idx >= BATCH * ne) return;
    int b = idx / ne;
    int e = idx - b * ne;
    int s = esrc[e];
    int d = edst[e];
    const float* vs = v + ((size_t)b * NV + s) * 3;
    __builtin_prefetch(vs, 0, 0);                     // -> global_prefetch_b8
    float* nd = nbr + ((size_t)b * NV + d) * 3;
    atomicAdd(&nd[0], vs[0]);
    atomicAdd(&nd[1], vs[1]);
    atomicAdd(&nd[2], vs[2]);
}

// ---------------------------------------------------------------------------
// Per-vertex loss terms + block LDS reduction to per-block partials.
// ---------------------------------------------------------------------------
__global__ __launch_bounds__(THREADS)
void loss_kernel(const float* __restrict__ v, const float* __restrict__ vn,
                 const float* __restrict__ nbr, const float* __restrict__ target,
                 const float* __restrict__ degree,
                 float* __restrict__ pp, float* __restrict__ pn, float* __restrict__ pl) {
    __shared__ float sh[3 * THREADS];
    int tid = threadIdx.x;
    int idx = blockIdx.x * THREADS + tid;             // over BATCH*NV
    float s_pos = 0.0f, s_nor = 0.0f, s_lap = 0.0f;
    if (idx < BATCH * NV) {
        int b = idx / NV;
        int i = idx - b * NV;
        const float* vv  = v   + (size_t)idx * 3;
        const float* vnv = vn  + (size_t)idx * 3;
        const float* nb  = nbr + (size_t)idx * 3;
        const float* t   = target + (size_t)b * 9 * NV + i;  // t[c*NV] = channel c
        // l_pos
#pragma unroll
        for (int c = 0; c < 3; ++c) {
            float d = vv[c] - t[(size_t)c * NV];
            s_pos += d * d;
        }
        // l_nor: normalize vn, then cos against tn with second clamp
        float n0 = vnv[0], n1 = vnv[1], n2 = vnv[2];
        float len  = sqrtf(n0 * n0 + n1 * n1 + n2 * n2);
        float inv  = 1.0f / fmaxf(len, 1e-12f);
        float u0 = n0 * inv, u1 = n1 * inv, u2 = n2 * inv;
        float lenu = sqrtf(u0 * u0 + u1 * u1 + u2 * u2);
        float t0 = t[(size_t)3 * NV], t1 = t[(size_t)4 * NV], t2 = t[(size_t)5 * NV];
        float lent = sqrtf(t0 * t0 + t1 * t1 + t2 * t2);
        float dp   = u0 * t0 + u1 * t1 + u2 * t2;
        float cosv = dp / fmaxf(lenu * lent, 1e-8f);
        s_nor = 1.0f - cosv;
        // l_lap
        float invdeg = 1.0f / degree[i];
#pragma unroll
        for (int c = 0; c < 3; ++c) {
            float lap = vv[c] - nb[c] * invdeg;
            float d = lap - t[(size_t)(6 + c) * NV];
            s_lap += d * d;
        }
    }
    sh[tid]               = s_pos;
    sh[THREADS + tid]     = s_nor;
    sh[2 * THREADS + tid] = s_lap;
    __syncthreads();
    for (int off = THREADS / 2; off > 0; off >>= 1) {
        if (tid < off) {
            sh[tid]               += sh[tid + off];
            sh[THREADS + tid]     += sh[THREADS + tid + off];
            sh[2 * THREADS + tid] += sh[2 * THREADS + tid + off];
        }
        __syncthreads();
    }
    if (tid == 0) {
        pp[blockIdx.x] = sh[0];
        pn[blockIdx.x] = sh[THREADS];
        pl[blockIdx.x] = sh[2 * THREADS];
    }
}

// ---------------------------------------------------------------------------
// Final reduction: one wave32. Sum NB64 partials per array with
// V_WMMA_F32_16X16X4_F32 accumulating in C: D = A x ones(4x16) + C.
// Sum of all 64 A entries == sum over the 16 rows of D (any column).
// C/D layout: vgpr r holds row m=r on lanes 0-15 and m=r+8 on lanes 16-31,
// so total = sum_r c[r] on lane 0 + sum_r c[r] on lane 16.
// ---------------------------------------------------------------------------
__global__ __launch_bounds__(32)
void finalize_kernel(const float* __restrict__ pp, const float* __restrict__ pn,
                     const float* __restrict__ pl, float* __restrict__ out) {
    const int lane = threadIdx.x;                     // 0..31, EXEC all ones
    v2f ones; ones.x = 1.0f; ones.y = 1.0f;
    v8f c0 = {}, c1 = {}, c2 = {};
    for (int base = 0; base < NB64; base += 64) {
        v2f a0, a1, a2;
        a0.x = pp[base + lane]; a0.y = pp[base + 32 + lane];
        a1.x = pn[base + lane]; a1.y = pn[base + 32 + lane];
        a2.x = pl[base + lane]; a2.y = pl[base + 32 + lane];
        c0 = __builtin_amdgcn_wmma_f32_16x16x4_f32(false, a0, false, ones,
                                                   (short)0, c0, false, false);
        c1 = __builtin_amdgcn_wmma_f32_16x16x4_f32(false, a1, false, ones,
                                                   (short)0, c1, false, false);
        c2 = __builtin_amdgcn_wmma_f32_16x16x4_f32(false, a2, false, ones,
                                                   (short)0, c2, false, false);
    }
    float s0 = 0.0f, s1 = 0.0f, s2 = 0.0f;
#pragma unroll
    for (int r = 0; r < 8; ++r) { s0 += c0[r]; s1 += c1[r]; s2 += c2[r]; }
    float t0 = __shfl(s0, 0, 32) + __shfl(s0, 16, 32);
    float t1 = __shfl(s1, 0, 32) + __shfl(s1, 16, 32);
    float t2 = __shfl(s2, 0, 32) + __shfl(s2, 16, 32);
    if (lane == 0) {
        const float n3 = (float)BATCH * (float)NV * 3.0f;
        const float n1 = (float)BATCH * (float)NV;
        out[0] = t0 / n3 + t1 / n1 + t2 / n3;   // FACTOR_* all 1.0
    }
}

// ---------------------------------------------------------------------------
// Host entry
// ---------------------------------------------------------------------------
extern "C" void kernel_launch(void* const* d_in, const int* in_sizes, int n_in,
                              void* d_out, int out_size, void* d_ws, size_t ws_size,
                              hipStream_t stream) {
    const float* inputs = (const float*)d_in[0];   // (B,3,GY,GX)
    const float* target = (const float*)d_in[1];   // (B,9,NV)
    const int*   faces  = (const int*)d_in[2];     // (NF,3)
    const int*   esrc   = (const int*)d_in[3];     // (NE,)
    const int*   edst   = (const int*)d_in[4];     // (NE,)
    const float* degree = (const float*)d_in[5];   // (NV,1)

    const int nf = in_sizes[2] / 3;
    const int ne = in_sizes[3];

    float* ws  = (float*)d_ws;
    float* v   = ws;                 // SZ floats
    float* vn  = ws + (size_t)SZ;    // SZ floats (zeroed)
    float* nbr = ws + (size_t)2 * SZ;// SZ floats (zeroed)
    float* pp  = ws + (size_t)3 * SZ;       // NB64 (zeroed incl. pad)
    float* pn  = pp + NB64;
    float* pl  = pn + NB64;

    // 1) zero vn, nbr_sum, and partial arrays (contiguous region)
    int zn = 2 * SZ + 3 * NB64;
    zero_kernel<<<(zn + THREADS - 1) / THREADS, THREADS, 0, stream>>>(vn, zn);

    // 2) build vertex positions (incl. pole means)
    build_v_kernel<<<(BATCH * NV + THREADS - 1) / THREADS, THREADS, 0, stream>>>(inputs, v);

    // 3) face-normal scatter
    face_kernel<<<(BATCH * nf + THREADS - 1) / THREADS, THREADS, 0, stream>>>(faces, v, vn, nf);

    // 4) Laplacian neighbor-sum scatter
    edge_kernel<<<(BATCH * ne + THREADS - 1) / THREADS, THREADS, 0, stream>>>(esrc, edst, v, nbr, ne);

    // 5) per-vertex loss terms -> per-block partials
    loss_kernel<<<NBLOCKS_LOSS, THREADS, 0, stream>>>(v, vn, nbr, target, degree, pp, pn, pl);

    // 6) one-wave WMMA reduction to the scalar loss
    finalize_kernel<<<1, 32, 0, stream>>>(pp, pn, pl, (float*)d_out);
}